// DelayAndSumLinear_31018253811715
// MI455X (gfx1250) — compile-verified
//
#include <hip/hip_runtime.h>
#include <stdint.h>

#define B_SZ   8
#define N_DET  128
#define N_T    2048
#define NPIX   (256 * 256)
#define NTILES (NPIX / 16)   // 4096 tiles of 16 pixels
#define WAVES  2             // waves per block (64 threads)
#define KROW   132           // padded int stride per pixel row (k0/alpha) -> conflict-free banks
#define VROW   144           // padded byte stride per pixel row (valid), 16B aligned

#if defined(__has_builtin)
#  if __has_builtin(__builtin_amdgcn_global_load_async_to_lds_b128)
#    define HAVE_ASYNC_B128 1
#  endif
#  if __has_builtin(__builtin_amdgcn_s_wait_asynccnt)
#    define HAVE_WAIT_ASYNC 1
#  endif
#  if __has_builtin(__builtin_amdgcn_wmma_f32_16x16x4_f32)
#    define HAVE_WMMA4 1
#  endif
#endif

typedef __attribute__((ext_vector_type(2))) float v2f;
typedef __attribute__((ext_vector_type(8))) float v8f;
typedef __attribute__((ext_vector_type(4))) int   i4;

// Exact parameter types for the async-copy builtin (from the round-1
// diagnostic: param 0 is '__device__ int __vector(4) *', i.e. addrspace(1)
// pointer to a 16-byte int vector; LDS side is the addrspace(3) twin).
typedef int v4i_t __attribute__((vector_size(16)));
typedef __attribute__((address_space(1))) v4i_t* g_v4i_p;
typedef __attribute__((address_space(3))) v4i_t* l_v4i_p;

// 16-byte global -> LDS copy. Async DMA path on gfx1250 (ASYNCcnt tracked),
// plain load/store fallback otherwise. LDS pointer is converted via integer
// truncation: flat addr[31:0] IS the LDS offset per the aperture rules.
__device__ __forceinline__ void cp16_to_lds(const void* g, void* l) {
#if defined(HAVE_ASYNC_B128)
  __builtin_amdgcn_global_load_async_to_lds_b128(
      (g_v4i_p)(uintptr_t)g, (l_v4i_p)(uint32_t)(uintptr_t)l, 0, 0);
#else
  *(i4*)l = *(const i4*)g;
#endif
}

__device__ __forceinline__ void wait_async_copies() {
#if defined(HAVE_ASYNC_B128)
#  if defined(HAVE_WAIT_ASYNC)
  __builtin_amdgcn_s_wait_asynccnt(0);
#  else
  asm volatile("s_wait_asynccnt 0" ::: "memory");
#  endif
#endif
}

// ---------------------------------------------------------------------------
// Kernel 1: per-batch mean and 1/std over (n_det, n_t). One block per batch.
// ---------------------------------------------------------------------------
__global__ __launch_bounds__(256) void stats_kernel(const float* __restrict__ sino,
                                                    float* __restrict__ stats) {
  __shared__ float ss[256];
  __shared__ float sq[256];
  const int b = blockIdx.x;
  const float* p = sino + (size_t)b * (N_DET * N_T);
  float s = 0.f, q = 0.f;
  for (int i = threadIdx.x; i < N_DET * N_T; i += 256) {
    float x = p[i];
    s += x;
    q = fmaf(x, x, q);
  }
  ss[threadIdx.x] = s;
  sq[threadIdx.x] = q;
  __syncthreads();
  for (int o = 128; o > 0; o >>= 1) {
    if (threadIdx.x < o) {
      ss[threadIdx.x] += ss[threadIdx.x + o];
      sq[threadIdx.x] += sq[threadIdx.x + o];
    }
    __syncthreads();
  }
  if (threadIdx.x == 0) {
    const float invN = 1.0f / (float)(N_DET * N_T);
    float mean = ss[0] * invN;
    float var  = fmaxf(sq[0] * invN - mean * mean, 0.f);
    float sd   = sqrtf(var + 1.1920929e-07f);  // finfo(f32).eps
    stats[2 * b]     = mean;
    stats[2 * b + 1] = 1.0f / sd;
  }
}

// ---------------------------------------------------------------------------
// Kernel 2: delay-and-sum. One wave per 16-pixel tile:
//   lane = (h, n): h = lane>>4 selects K-half, n = lane&15 is the pixel column.
//   For each chunk of 4 detectors (K=4) and each batch b: gather+interp into
//   B-operand, apod weights into A-operand, accumulate via WMMA f32 16x16x4.
//   D row 0 (VGPR0, lanes 0..15) holds the apod-weighted detector sum.
// ---------------------------------------------------------------------------
__global__ __launch_bounds__(64) void das_kernel(
    const float* __restrict__ sino, const float* __restrict__ alpha,
    const float* __restrict__ apod, const int* __restrict__ k0,
    const unsigned char* __restrict__ valid, const float* __restrict__ stats,
    float* __restrict__ out) {
  __shared__ int           sK [WAVES][16 * KROW];
  __shared__ float         sAl[WAVES][16 * KROW];
  __shared__ unsigned char sVl[WAVES][16 * VROW];
  __shared__ float         sAp[N_DET];

  const int tid  = threadIdx.x;
  const int wave = tid >> 5;
  const int lane = tid & 31;
  const int h    = lane >> 4;  // which half of the wave (K offset 0 / 2)
  const int n    = lane & 15;  // pixel column in tile

  for (int i = tid; i < N_DET; i += blockDim.x) sAp[i] = apod[i];
  __syncthreads();

  float norm = 0.f;
  for (int i = 0; i < N_DET; ++i) norm += sAp[i];
  const float invnorm = 1.0f / fmaxf(norm, 1.17549435e-38f);  // finfo(f32).tiny

  float mean[B_SZ], istd[B_SZ];
#pragma unroll
  for (int b = 0; b < B_SZ; ++b) {
    mean[b] = stats[2 * b];
    istd[b] = stats[2 * b + 1];
  }

  for (int tile = blockIdx.x * WAVES + wave; tile < NTILES;
       tile += gridDim.x * WAVES) {
    const int p0 = tile * 16;
    const int*           gK = k0    + p0 * N_DET;  // 8KB contiguous
    const float*         gA = alpha + p0 * N_DET;  // 8KB contiguous
    const unsigned char* gV = valid + p0 * N_DET;  // 2KB contiguous

    // Stage the wave's 16x128 LUT tile into its private LDS slice (async DMA),
    // scattering into padded rows for bank-conflict-free reads.
    for (int c = lane; c < 512; c += 32) {
      int pix = c >> 5, off = (c & 31) << 2;
      cp16_to_lds(gK + pix * N_DET + off, &sK[wave][pix * KROW + off]);
    }
    for (int c = lane; c < 512; c += 32) {
      int pix = c >> 5, off = (c & 31) << 2;
      cp16_to_lds(gA + pix * N_DET + off, &sAl[wave][pix * KROW + off]);
    }
    for (int c = lane; c < 128; c += 32) {
      int pix = c >> 3, off = (c & 7) << 4;
      cp16_to_lds(gV + pix * N_DET + off, &sVl[wave][pix * VROW + off]);
    }
    wait_async_copies();

#if defined(HAVE_WMMA4)
    v8f acc[B_SZ];
    v8f zero = {};
#pragma unroll
    for (int b = 0; b < B_SZ; ++b) acc[b] = zero;
#else
    float accf[B_SZ];
#pragma unroll
    for (int b = 0; b < B_SZ; ++b) accf[b] = 0.f;
#endif

    for (int kc = 0; kc < N_DET / 4; ++kc) {
      // This lane supplies K = 2h and K = 2h+1, i.e. detectors d0, d0+1.
      const int d0 = kc * 4 + 2 * h;
      const int d1 = d0 + 1;
      const int   ka = sK [wave][n * KROW + d0];
      const int   kb = sK [wave][n * KROW + d1];
      const float aa = sAl[wave][n * KROW + d0];
      const float ab = sAl[wave][n * KROW + d1];
      const float va = sVl[wave][n * VROW + d0] ? 1.0f : 0.0f;
      const float vb = sVl[wave][n * VROW + d1] ? 1.0f : 0.0f;
      const float w0 = sAp[d0];
      const float w1 = sAp[d1];
      const int base0 = d0 * N_T + ka;
      const int base1 = d1 * N_T + kb;
#if defined(HAVE_WMMA4)
      v2f A;  // A[m][K] = apod[kc*4+K], identical for every row m
      A.x = w0;
      A.y = w1;
#endif
#pragma unroll
      for (int b = 0; b < B_SZ; ++b) {
        const float* Sb = sino + b * (N_DET * N_T);
        // L2-resident scattered gathers (sino is 8MB, L2 is 192MB).
        float x0 = Sb[base0], x1 = Sb[base0 + 1];
        float y0 = Sb[base1], y1 = Sb[base1 + 1];
        float ra = (1.0f - aa) * x0 + aa * x1;
        float rb = (1.0f - ab) * y0 + ab * y1;
        // on-the-fly normalization: ((1-a)s0+a s1) == (r - mean) * istd
        float ska = va * istd[b] * (ra - mean[b]);
        float skb = vb * istd[b] * (rb - mean[b]);
#if defined(HAVE_WMMA4)
        v2f Bv;  // B[K][n] = sk(det kc*4+K, pixel n)
        Bv.x = ska;
        Bv.y = skb;
        acc[b] = __builtin_amdgcn_wmma_f32_16x16x4_f32(
            false, A, false, Bv, (short)0, acc[b], false, false);
#else
        accf[b] += w0 * ska + w1 * skb;
#endif
      }
    }

    float res[B_SZ];
#pragma unroll
    for (int b = 0; b < B_SZ; ++b) {
#if defined(HAVE_WMMA4)
      res[b] = acc[b][0];  // D row 0: M=0, N=lane (lanes 0..15)
#else
      res[b] = accf[b] + __shfl_xor(accf[b], 16, 32);
#endif
    }
    if (lane < 16) {
#pragma unroll
      for (int b = 0; b < B_SZ; ++b)
        out[b * NPIX + p0 + n] = res[b] * invnorm;
    }
  }
}

extern "C" void kernel_launch(void* const* d_in, const int* in_sizes, int n_in,
                              void* d_out, int out_size, void* d_ws,
                              size_t ws_size, hipStream_t stream) {
  (void)in_sizes; (void)n_in; (void)out_size; (void)ws_size;
  const float*         sino  = (const float*)d_in[0];
  const float*         alpha = (const float*)d_in[1];
  const float*         apod  = (const float*)d_in[2];
  const int*           k0    = (const int*)d_in[3];
  const unsigned char* valid = (const unsigned char*)d_in[4];
  float* out   = (float*)d_out;
  float* stats = (float*)d_ws;  // 16 floats: (mean, 1/std) per batch

  stats_kernel<<<B_SZ, 256, 0, stream>>>(sino, stats);
  das_kernel<<<NTILES / WAVES, 64, 0, stream>>>(sino, alpha, apod, k0, valid,
                                                stats, out);
}